// TernaryConv1D_17892833755666
// MI455X (gfx1250) — compile-verified
//
#include <hip/hip_runtime.h>
#include <hip/hip_bf16.h>

typedef __attribute__((ext_vector_type(2))) float v2f;
typedef __attribute__((ext_vector_type(8))) float v8f;

#define L_DIM   8192
#define F_DIM   256
#define K_TAPS  16
#define B_DIM   32

// ---------------------------------------------------------------------------
// Kernel 1: ternary-quantize the weights.
// alpha = mean(|w|) over all 16*256 elements; thr = 0.7*alpha;
// wq[k][f] = (|w|<thr) ? 0 : copysign(alpha, w).  Output: 16KB f32 table in ws.
// ---------------------------------------------------------------------------
__global__ void ternary_quantize_kernel(const float* __restrict__ w,
                                        float* __restrict__ wq) {
    __shared__ float red[256];
    const int f = threadIdx.x;           // 256 threads, one per output channel
    float s = 0.0f;
#pragma unroll
    for (int k = 0; k < K_TAPS; ++k) s += fabsf(w[k * F_DIM + f]);
    red[f] = s;
    __syncthreads();
#pragma unroll
    for (int off = 128; off > 0; off >>= 1) {
        if (f < off) red[f] += red[f + off];
        __syncthreads();
    }
    const float alpha = red[0] * (1.0f / (K_TAPS * F_DIM));
    const float thr   = 0.7f * alpha;
#pragma unroll
    for (int k = 0; k < K_TAPS; ++k) {
        const float v = w[k * F_DIM + f];
        wq[k * F_DIM + f] = (fabsf(v) < thr) ? 0.0f : copysignf(alpha, v);
    }
}

// ---------------------------------------------------------------------------
// Kernel 2: conv as GEMM via V_WMMA_F32_16X16X4_F32.
// Each wave: one 16-row output strip x all 256 channels (16 N-tiles).
// A (16x4 f32, per ISA layout): lanes 0-15 M=0..15; VGPR0=K{0|2}, VGPR1=K{1|3}.
// C/D (16x16 f32): VGPR r: lanes 0-15 -> (M=r,N=lane); lanes 16-31 -> (M=r+8).
// ---------------------------------------------------------------------------
__global__ void __launch_bounds__(256, 2)
conv_wmma_kernel(const float* __restrict__ x,
                 const float* __restrict__ wq,
                 const float* __restrict__ bias,
                 float* __restrict__ out) {
    const int lane = threadIdx.x & 31;
    const int wave = threadIdx.x >> 5;
    const int m    = lane & 15;          // M index within tile
    const int half = lane >> 4;          // 0: lanes 0-15, 1: lanes 16-31

    const long long waveId = (long long)blockIdx.x * 8 + wave;
    const long long l0     = waveId * 16;           // global output row of strip
    const int  pos0        = (int)(l0 & (L_DIM - 1));
    const float* xrow      = x + (l0 - pos0);       // base of this batch's row

    // ---- A fragments: X[m, k] = x[pos0 + m + k - 7], 4 k-blocks of 4 ----
    // Branch-free SAME padding: clamp address, load unconditionally (always
    // in-bounds within the batch row), then cndmask the value to zero.
    v2f a[4];
#pragma unroll
    for (int kb = 0; kb < 4; ++kb) {
        const int k0 = 4 * kb + 2 * half;           // VGPR0 holds K = {0,2}+4kb
        const int p0 = pos0 + m + k0 - 7;
        const int p1 = p0 + 1;
        const int p0c = min(max(p0, 0), L_DIM - 1);
        const int p1c = min(max(p1, 0), L_DIM - 1);
        const float v0 = xrow[p0c];
        const float v1 = xrow[p1c];
        a[kb].x = ((unsigned)p0 < (unsigned)L_DIM) ? v0 : 0.0f;
        a[kb].y = ((unsigned)p1 < (unsigned)L_DIM) ? v1 : 0.0f;
    }

    // ---- sweep the 16 N-tiles; A stays in registers ----
    // unroll 4: four independent C accumulators so the L2-hot weight loads,
    // the WMMAs, and the NT store stream overlap.
#pragma unroll 4
    for (int nt = 0; nt < 16; ++nt) {
        const int col = nt * 16 + m;                // channel index for this lane
        const float bv = bias[col];
        v8f c = { bv, bv, bv, bv, bv, bv, bv, bv };

#pragma unroll
        for (int kb = 0; kb < 4; ++kb) {
            const int k0 = 4 * kb + 2 * half;       // B: VGPR0 rows K{0|2}+4kb
            v2f bb;
            bb.x = wq[(k0    ) * F_DIM + col];
            bb.y = wq[(k0 + 1) * F_DIM + col];
            // 8 args: (neg_a, A, neg_b, B, c_mod, C, reuse_a, reuse_b)
            c = __builtin_amdgcn_wmma_f32_16x16x4_f32(
                    false, a[kb], false, bb, (short)0, c, false, false);
        }

        // ---- stream the 16x16 tile out with non-temporal stores ----
        float* orow = out + (l0 + (long long)(half * 8)) * F_DIM + col;
#pragma unroll
        for (int r = 0; r < 8; ++r) {
            __builtin_nontemporal_store(c[r], orow + (long long)r * F_DIM);
        }
    }
}

// ---------------------------------------------------------------------------
// Launch: quantize weights into ws, then the WMMA conv.
// ---------------------------------------------------------------------------
extern "C" void kernel_launch(void* const* d_in, const int* in_sizes, int n_in,
                              void* d_out, int out_size, void* d_ws, size_t ws_size,
                              hipStream_t stream) {
    const float* x = (const float*)d_in[0];   // (32, 8192, 1)
    const float* w = (const float*)d_in[1];   // (16, 1, 256)
    const float* b = (const float*)d_in[2];   // (256,)
    float* out = (float*)d_out;               // (32, 8192, 256)
    float* wq  = (float*)d_ws;                // 16*256 f32 = 16 KB scratch

    ternary_quantize_kernel<<<1, 256, 0, stream>>>(w, wq);

    const long long total_rows = (long long)B_DIM * L_DIM;   // 262144
    const int waves  = (int)(total_rows / 16);                // 16384
    const int blocks = waves / 8;                             // 2048
    conv_wmma_kernel<<<blocks, 256, 0, stream>>>(x, wq, b, out);
}